// FullGraphGNN_27169963114791
// MI455X (gfx1250) — compile-verified
//
#include <hip/hip_runtime.h>
#include <hip/hip_bf16.h>

typedef float v2f __attribute__((ext_vector_type(2)));
typedef float v8f __attribute__((ext_vector_type(8)));

#define LRELU(v) ((v) > 0.0f ? (v) : 0.2f * (v))

__device__ inline float waveReduceSum(float v) {
#pragma unroll
  for (int o = 16; o > 0; o >>= 1) v += __shfl_xor(v, o, 32);
  return v;
}

__device__ inline void atomicMaxF(float* addr, float v) {
  if (v >= 0.0f) atomicMax((int*)addr, __float_as_int(v));
  else           atomicMin((unsigned int*)addr, __float_as_uint(v));
}

// ---------------------------------------------------------------------------
// GEMM: D[M x Nc] = A[M x K] @ W[K x Nc] (+ bias[Nc]) (+ Cin[M x Nc])
// One wave computes a 16-row x 128-col strip using V_WMMA_F32_16X16X4_F32.
// Requires M % 16 == 0, K % 4 == 0, Nc % 128 == 0 (true for all call sites).
// ---------------------------------------------------------------------------
__global__ __launch_bounds__(128)
void gemm_wmma_f32(const float* __restrict__ A, const float* __restrict__ Wm,
                   const float* __restrict__ bias, const float* __restrict__ Cin,
                   float* __restrict__ D, int M, int K, int Nc, int colBlocks) {
  const int lane = threadIdx.x & 31;
  const int wIB  = threadIdx.x >> 5;
  const long wave = (long)blockIdx.x * 4 + wIB;
  const long rowTiles = (long)(M >> 4);
  const long totalWaves = rowTiles * colBlocks;
  if (wave >= totalWaves) return;                  // wave-uniform: EXEC stays full

  const int cb = (int)(wave % colBlocks);
  const int rt = (int)(wave / colBlocks);
  const int m0 = rt << 4;
  const int c0 = cb << 7;                          // 128 columns per wave
  const int hi  = lane >> 4;                       // 0: lanes 0-15, 1: lanes 16-31
  const int l16 = lane & 15;
  const int m   = m0 + l16;                        // A row for this lane
  const int rowOff = hi << 3;                      // C/D: hi half holds rows +8..+15

  v8f acc[8];
#pragma unroll
  for (int t = 0; t < 8; ++t) {
    const int n = c0 + (t << 4) + l16;
    const float binit = bias ? bias[n] : 0.0f;
    if (Cin) {
#pragma unroll
      for (int j = 0; j < 8; ++j)
        acc[t][j] = Cin[(long)(m0 + rowOff + j) * Nc + n] + binit;
    } else {
#pragma unroll
      for (int j = 0; j < 8; ++j) acc[t][j] = binit;
    }
  }

  const float* Arow = A + (long)m * K;
  for (int k = 0; k < K; k += 4) {
    const int k0 = k + (hi << 1);                  // lanes0-15: k,k+1 ; lanes16-31: k+2,k+3
    v2f a;
    a.x = Arow[k0];
    a.y = Arow[k0 + 1];
#pragma unroll
    for (int t = 0; t < 8; ++t) {
      const int n = c0 + (t << 4) + l16;
      v2f b;
      b.x = Wm[(long)k0 * Nc + n];
      b.y = Wm[(long)(k0 + 1) * Nc + n];
      acc[t] = __builtin_amdgcn_wmma_f32_16x16x4_f32(false, a, false, b,
                                                     (short)0, acc[t], false, false);
    }
  }

#pragma unroll
  for (int t = 0; t < 8; ++t) {
    const int n = c0 + (t << 4) + l16;
#pragma unroll
    for (int j = 0; j < 8; ++j)
      D[(long)(m0 + rowOff + j) * Nc + n] = acc[t][j];
  }
}

// ---------------------------------------------------------------------------
// GAT per-node attention scores: es[i]=h[i]·a_s, ed[i]=h[i]·a_d. Wave per node.
// ---------------------------------------------------------------------------
__global__ __launch_bounds__(256)
void gat_scores(const float* __restrict__ h, const float* __restrict__ as,
                const float* __restrict__ ad, float* __restrict__ es,
                float* __restrict__ ed, int n) {
  const int lane = threadIdx.x & 31;
  const long i = ((long)blockIdx.x * blockDim.x + threadIdx.x) >> 5;
  if (i >= n) return;
  const float4 hv = *(const float4*)(h + i * 128 + lane * 4);
  const float4 av = *(const float4*)(as + lane * 4);
  const float4 dv = *(const float4*)(ad + lane * 4);
  float s = hv.x * av.x + hv.y * av.y + hv.z * av.z + hv.w * av.w;
  float d = hv.x * dv.x + hv.y * dv.y + hv.z * dv.z + hv.w * dv.w;
  s = waveReduceSum(s);
  d = waveReduceSum(d);
  if (lane == 0) { es[i] = s; ed[i] = d; }
}

// emax seeded with the self-loop score. Thread per node.
__global__ __launch_bounds__(256)
void emax_init(const float* __restrict__ es, const float* __restrict__ ed,
               float* __restrict__ emax, int n) {
  const long i = (long)blockIdx.x * blockDim.x + threadIdx.x;
  if (i >= n) return;
  const float v = es[i] + ed[i];
  emax[i] = LRELU(v);
}

// Edge pass A: scatter-max of edge scores. Thread per edge.
__global__ __launch_bounds__(256)
void edge_max(const long long* __restrict__ ei, const float* __restrict__ es,
              const float* __restrict__ ed, float* __restrict__ emax, long nE) {
  const long e = (long)blockIdx.x * blockDim.x + threadIdx.x;
  if (e >= nE) return;
  const int s = (int)ei[e];
  const int d = (int)ei[nE + e];
  float v = es[s] + ed[d];
  v = LRELU(v);
  atomicMaxF(&emax[d], v);
}

// Self-loop contribution: denom[i]=w_self, x1[i][:]=w_self*h[i][:]. Wave per node.
__global__ __launch_bounds__(256)
void gat_self_init(const float* __restrict__ es, const float* __restrict__ ed,
                   const float* __restrict__ emax, const float* __restrict__ h,
                   float* __restrict__ denom, float* __restrict__ x1, int n) {
  const int lane = threadIdx.x & 31;
  const long i = ((long)blockIdx.x * blockDim.x + threadIdx.x) >> 5;
  if (i >= n) return;
  float v = es[i] + ed[i];
  v = LRELU(v);
  const float w = __expf(v - emax[i]);
  if (lane == 0) denom[i] = w;
  const float4 hv = *(const float4*)(h + i * 128 + lane * 4);
  float4 o;
  o.x = w * hv.x; o.y = w * hv.y; o.z = w * hv.z; o.w = w * hv.w;
  *(float4*)(x1 + i * 128 + lane * 4) = o;
}

// Edge pass B (fused, heavy): GAT numerator + denom + SAGE1 sum + degree.
// One wave per edge; each lane owns 4 of the 128 feature columns.
__global__ __launch_bounds__(256)
void edge_agg1(const long long* __restrict__ ei, const float* __restrict__ es,
               const float* __restrict__ ed, const float* __restrict__ emax,
               const float* __restrict__ h, const float* __restrict__ x,
               float* __restrict__ denom, float* __restrict__ deg,
               float* __restrict__ x1, float* __restrict__ ag1, long nE) {
  const int lane = threadIdx.x & 31;
  const long e = ((long)blockIdx.x * blockDim.x + threadIdx.x) >> 5;
  if (e >= nE) return;
  const int s = (int)ei[e];
  const int d = (int)ei[nE + e];
  float v = es[s] + ed[d];
  v = LRELU(v);
  const float w = __expf(v - emax[d]);
  if (lane == 0) { atomicAdd(denom + d, w); atomicAdd(deg + d, 1.0f); }

  const float4 hv = *(const float4*)(h + (long)s * 128 + lane * 4);
  float* xo = x1 + (long)d * 128 + lane * 4;
  atomicAdd(xo + 0, w * hv.x); atomicAdd(xo + 1, w * hv.y);
  atomicAdd(xo + 2, w * hv.z); atomicAdd(xo + 3, w * hv.w);

  const float4 xv = *(const float4*)(x + (long)s * 128 + lane * 4);
  float* ao = ag1 + (long)d * 128 + lane * 4;
  atomicAdd(ao + 0, xv.x); atomicAdd(ao + 1, xv.y);
  atomicAdd(ao + 2, xv.z); atomicAdd(ao + 3, xv.w);
}

// Finalize layer-1 aggregations: x1 = x1/denom + b_gat ; agg1 /= max(deg,1).
__global__ __launch_bounds__(256)
void finalize1(float* __restrict__ x1, float* __restrict__ ag1,
               const float* __restrict__ denom, const float* __restrict__ deg,
               const float* __restrict__ bg, int n) {
  const int lane = threadIdx.x & 31;
  const long i = ((long)blockIdx.x * blockDim.x + threadIdx.x) >> 5;
  if (i >= n) return;
  const float invD = 1.0f / denom[i];
  const float invG = 1.0f / fmaxf(deg[i], 1.0f);
  float4 a = *(const float4*)(x1 + i * 128 + lane * 4);
  const float4 b = *(const float4*)(bg + lane * 4);
  a.x = a.x * invD + b.x; a.y = a.y * invD + b.y;
  a.z = a.z * invD + b.z; a.w = a.w * invD + b.w;
  *(float4*)(x1 + i * 128 + lane * 4) = a;
  float4 g = *(const float4*)(ag1 + i * 128 + lane * 4);
  g.x *= invG; g.y *= invG; g.z *= invG; g.w *= invG;
  *(float4*)(ag1 + i * 128 + lane * 4) = g;
}

// LN1: hcat[i] = relu(LN([x1[i] | x2[i]]) * g + b). Wave per node, 8 elems/lane.
__global__ __launch_bounds__(256)
void ln1_relu(const float* __restrict__ x1, const float* __restrict__ x2,
              const float* __restrict__ g, const float* __restrict__ b,
              float* __restrict__ hc, int n) {
  const int lane = threadIdx.x & 31;
  const long i = ((long)blockIdx.x * blockDim.x + threadIdx.x) >> 5;
  if (i >= n) return;
  const float4 a = *(const float4*)(x1 + i * 128 + lane * 4);
  const float4 c = *(const float4*)(x2 + i * 128 + lane * 4);
  float s = a.x + a.y + a.z + a.w + c.x + c.y + c.z + c.w;
  s = waveReduceSum(s);
  const float mu = s * (1.0f / 256.0f);
  float vv = 0.0f;
  {
    float t;
    t = a.x - mu; vv += t * t; t = a.y - mu; vv += t * t;
    t = a.z - mu; vv += t * t; t = a.w - mu; vv += t * t;
    t = c.x - mu; vv += t * t; t = c.y - mu; vv += t * t;
    t = c.z - mu; vv += t * t; t = c.w - mu; vv += t * t;
  }
  vv = waveReduceSum(vv);
  const float rs = rsqrtf(vv * (1.0f / 256.0f) + 1e-5f);
  const int c0 = lane * 4;
  const float4 g0 = *(const float4*)(g + c0);
  const float4 b0 = *(const float4*)(b + c0);
  const float4 g1 = *(const float4*)(g + 128 + c0);
  const float4 b1 = *(const float4*)(b + 128 + c0);
  float4 o0, o1;
  o0.x = fmaxf((a.x - mu) * rs * g0.x + b0.x, 0.0f);
  o0.y = fmaxf((a.y - mu) * rs * g0.y + b0.y, 0.0f);
  o0.z = fmaxf((a.z - mu) * rs * g0.z + b0.z, 0.0f);
  o0.w = fmaxf((a.w - mu) * rs * g0.w + b0.w, 0.0f);
  o1.x = fmaxf((c.x - mu) * rs * g1.x + b1.x, 0.0f);
  o1.y = fmaxf((c.y - mu) * rs * g1.y + b1.y, 0.0f);
  o1.z = fmaxf((c.z - mu) * rs * g1.z + b1.z, 0.0f);
  o1.w = fmaxf((c.w - mu) * rs * g1.w + b1.w, 0.0f);
  *(float4*)(hc + i * 256 + c0) = o0;
  *(float4*)(hc + i * 256 + 128 + c0) = o1;
}

// Edge pass C: agg2[dst] += hcat[src]. Wave per edge, 8 cols (two float4) per lane.
__global__ __launch_bounds__(256)
void edge_agg2(const long long* __restrict__ ei, const float* __restrict__ hc,
               float* __restrict__ ag2, long nE) {
  const int lane = threadIdx.x & 31;
  const long e = ((long)blockIdx.x * blockDim.x + threadIdx.x) >> 5;
  if (e >= nE) return;
  const int s = (int)ei[e];
  const int d = (int)ei[nE + e];
  const float4 h0 = *(const float4*)(hc + (long)s * 256 + lane * 8);
  const float4 h1 = *(const float4*)(hc + (long)s * 256 + lane * 8 + 4);
  float* o = ag2 + (long)d * 256 + lane * 8;
  atomicAdd(o + 0, h0.x); atomicAdd(o + 1, h0.y);
  atomicAdd(o + 2, h0.z); atomicAdd(o + 3, h0.w);
  atomicAdd(o + 4, h1.x); atomicAdd(o + 5, h1.y);
  atomicAdd(o + 6, h1.z); atomicAdd(o + 7, h1.w);
}

// agg2 /= max(deg, 1). Wave per node.
__global__ __launch_bounds__(256)
void agg2_mean(float* __restrict__ ag2, const float* __restrict__ deg, int n) {
  const int lane = threadIdx.x & 31;
  const long i = ((long)blockIdx.x * blockDim.x + threadIdx.x) >> 5;
  if (i >= n) return;
  const float invG = 1.0f / fmaxf(deg[i], 1.0f);
  float4 a = *(const float4*)(ag2 + i * 256 + lane * 8);
  float4 b = *(const float4*)(ag2 + i * 256 + lane * 8 + 4);
  a.x *= invG; a.y *= invG; a.z *= invG; a.w *= invG;
  b.x *= invG; b.y *= invG; b.z *= invG; b.w *= invG;
  *(float4*)(ag2 + i * 256 + lane * 8) = a;
  *(float4*)(ag2 + i * 256 + lane * 8 + 4) = b;
}

// LN2 + ReLU + classifier dot in one pass. Wave per node.
__global__ __launch_bounds__(256)
void ln2_cls(const float* __restrict__ h2, const float* __restrict__ g,
             const float* __restrict__ b, const float* __restrict__ wc,
             const float* __restrict__ bc, float* __restrict__ out, int n) {
  const int lane = threadIdx.x & 31;
  const long i = ((long)blockIdx.x * blockDim.x + threadIdx.x) >> 5;
  if (i >= n) return;
  const int c0 = lane * 8;
  const float4 a = *(const float4*)(h2 + i * 256 + c0);
  const float4 c = *(const float4*)(h2 + i * 256 + c0 + 4);
  float s = a.x + a.y + a.z + a.w + c.x + c.y + c.z + c.w;
  s = waveReduceSum(s);
  const float mu = s * (1.0f / 256.0f);
  float vv = 0.0f;
  {
    float t;
    t = a.x - mu; vv += t * t; t = a.y - mu; vv += t * t;
    t = a.z - mu; vv += t * t; t = a.w - mu; vv += t * t;
    t = c.x - mu; vv += t * t; t = c.y - mu; vv += t * t;
    t = c.z - mu; vv += t * t; t = c.w - mu; vv += t * t;
  }
  vv = waveReduceSum(vv);
  const float rs = rsqrtf(vv * (1.0f / 256.0f) + 1e-5f);
  const float4 g0 = *(const float4*)(g + c0);
  const float4 b0 = *(const float4*)(b + c0);
  const float4 g1 = *(const float4*)(g + c0 + 4);
  const float4 b1 = *(const float4*)(b + c0 + 4);
  const float4 w0 = *(const float4*)(wc + c0);
  const float4 w1 = *(const float4*)(wc + c0 + 4);
  float acc = 0.0f;
  acc += fmaxf((a.x - mu) * rs * g0.x + b0.x, 0.0f) * w0.x;
  acc += fmaxf((a.y - mu) * rs * g0.y + b0.y, 0.0f) * w0.y;
  acc += fmaxf((a.z - mu) * rs * g0.z + b0.z, 0.0f) * w0.z;
  acc += fmaxf((a.w - mu) * rs * g0.w + b0.w, 0.0f) * w0.w;
  acc += fmaxf((c.x - mu) * rs * g1.x + b1.x, 0.0f) * w1.x;
  acc += fmaxf((c.y - mu) * rs * g1.y + b1.y, 0.0f) * w1.y;
  acc += fmaxf((c.z - mu) * rs * g1.z + b1.z, 0.0f) * w1.z;
  acc += fmaxf((c.w - mu) * rs * g1.w + b1.w, 0.0f) * w1.w;
  acc = waveReduceSum(acc);
  if (lane == 0) out[i] = acc + bc[0];
}

// ---------------------------------------------------------------------------
extern "C" void kernel_launch(void* const* d_in, const int* in_sizes, int n_in,
                              void* d_out, int out_size, void* d_ws, size_t ws_size,
                              hipStream_t stream) {
  const float*     x    = (const float*)d_in[0];
  const long long* ei   = (const long long*)d_in[1];   // int64 edge_index (2,E)
  const float* W_gat = (const float*)d_in[2];
  const float* a_src = (const float*)d_in[3];
  const float* a_dst = (const float*)d_in[4];
  const float* b_gat = (const float*)d_in[5];
  const float* W1l   = (const float*)d_in[6];
  const float* b1l   = (const float*)d_in[7];
  const float* W1r   = (const float*)d_in[8];
  const float* ln1g  = (const float*)d_in[9];
  const float* ln1b  = (const float*)d_in[10];
  const float* W2l   = (const float*)d_in[11];
  const float* b2l   = (const float*)d_in[12];
  const float* W2r   = (const float*)d_in[13];
  const float* ln2g  = (const float*)d_in[14];
  const float* ln2b  = (const float*)d_in[15];
  const float* Wc    = (const float*)d_in[16];
  const float* bc    = (const float*)d_in[17];
  float* out = (float*)d_out;

  const long N  = (long)in_sizes[0] / 128;
  const long nE = (long)in_sizes[1] / 2;

  // Workspace bump allocation (floats). Aliasing:
  //   f_h  : h = x@W_gat, reused later as x2
  //   f_x1 : GAT output; f_x1..f_x1+256N (x1+agg1) reused as agg2 after LN1
  float* Wp = (float*)d_ws;
  size_t p = 0;
  float* f_h   = Wp + p; p += (size_t)N * 128;   // h_gat -> x2
  float* f_x1  = Wp + p; p += (size_t)N * 128;   // x1   -> agg2 (first half)
  float* f_ag1 = Wp + p; p += (size_t)N * 128;   // agg1 -> agg2 (second half)
  float* f_hc  = Wp + p; p += (size_t)N * 256;   // hcat
  float* f_h2  = Wp + p; p += (size_t)N * 256;   // h2
  float* f_es   = Wp + p; p += (size_t)N;
  float* f_ed   = Wp + p; p += (size_t)N;
  float* f_emax = Wp + p; p += (size_t)N;
  float* f_den  = Wp + p; p += (size_t)N;
  float* f_deg  = Wp + p; p += (size_t)N;
  float* f_x2  = f_h;
  float* f_ag2 = f_x1;                           // 256N contiguous floats

  const int TPB = 256;
  const int nodeWaveBlocks = (int)((N * 32 + TPB - 1) / TPB);
  const int nodeThreadBlocks = (int)((N + TPB - 1) / TPB);
  const int edgeThreadBlocks = (int)((nE + TPB - 1) / TPB);
  const int edgeWaveBlocks = (int)((nE * 32 + TPB - 1) / TPB);

  auto gemm = [&](const float* A, const float* Wm, const float* bias,
                  const float* Cin, float* D, int M, int K, int Nc) {
    const int colBlocks = Nc / 128;
    const long waves = (long)(M / 16) * colBlocks;
    const int blocks = (int)((waves + 3) / 4);
    gemm_wmma_f32<<<blocks, 128, 0, stream>>>(A, Wm, bias, Cin, D, M, K, Nc, colBlocks);
  };

  // ---- Layer 1: GAT branch -------------------------------------------------
  gemm(x, W_gat, nullptr, nullptr, f_h, (int)N, 128, 128);            // h = x @ W_gat
  gat_scores<<<nodeWaveBlocks, TPB, 0, stream>>>(f_h, a_src, a_dst, f_es, f_ed, (int)N);
  emax_init<<<nodeThreadBlocks, TPB, 0, stream>>>(f_es, f_ed, f_emax, (int)N);
  edge_max<<<edgeThreadBlocks, TPB, 0, stream>>>(ei, f_es, f_ed, f_emax, nE);
  gat_self_init<<<nodeWaveBlocks, TPB, 0, stream>>>(f_es, f_ed, f_emax, f_h, f_den, f_x1, (int)N);

  hipMemsetAsync(f_ag1, 0, (size_t)N * 128 * sizeof(float), stream);
  hipMemsetAsync(f_deg, 0, (size_t)N * sizeof(float), stream);

  // Fused heavy edge pass: GAT numerator+denominator, SAGE1 sum, degree.
  edge_agg1<<<edgeWaveBlocks, TPB, 0, stream>>>(ei, f_es, f_ed, f_emax, f_h, x,
                                                f_den, f_deg, f_x1, f_ag1, nE);
  finalize1<<<nodeWaveBlocks, TPB, 0, stream>>>(f_x1, f_ag1, f_den, f_deg, b_gat, (int)N);

  // ---- Layer 1: SAGE linear part (x2 overwrites h, which is now dead) ------
  gemm(f_ag1, W1l, b1l, nullptr, f_x2, (int)N, 128, 128);             // x2 = agg1@W1l + b1l
  gemm(x, W1r, nullptr, f_x2, f_x2, (int)N, 128, 128);                // x2 += x@W1r

  ln1_relu<<<nodeWaveBlocks, TPB, 0, stream>>>(f_x1, f_x2, ln1g, ln1b, f_hc, (int)N);

  // ---- Layer 2: SAGE(hidden) ----------------------------------------------
  hipMemsetAsync(f_ag2, 0, (size_t)N * 256 * sizeof(float), stream);
  edge_agg2<<<edgeWaveBlocks, TPB, 0, stream>>>(ei, f_hc, f_ag2, nE);
  agg2_mean<<<nodeWaveBlocks, TPB, 0, stream>>>(f_ag2, f_deg, (int)N);

  gemm(f_ag2, W2l, b2l, nullptr, f_h2, (int)N, 256, 256);             // h2 = agg2@W2l + b2l
  gemm(f_hc, W2r, nullptr, f_h2, f_h2, (int)N, 256, 256);             // h2 += hcat@W2r

  ln2_cls<<<nodeWaveBlocks, TPB, 0, stream>>>(f_h2, ln2g, ln2b, Wc, bc, out, (int)N);
}